// DCMF_76201309766068
// MI455X (gfx1250) — compile-verified
//
#include <hip/hip_runtime.h>

// ---------------------------------------------------------------------------
// Problem constants (from the reference)
// ---------------------------------------------------------------------------
#define U_CNT 50000
#define I_CNT 30000
#define N_CNT (U_CNT + I_CNT)   // 80000
#define DE    64

typedef __attribute__((ext_vector_type(2))) float v2f;
typedef __attribute__((ext_vector_type(4))) float v4f;
typedef __attribute__((ext_vector_type(8))) float v8f;

// LDS row pad: 34 floats -> row base 136B (8B aligned, so v2f LDS ops become
// ds_load_b64/ds_store_b64), and (34*r + c) % 64 is conflict-free across the
// 16-lane row groups of a wave.
#define LPAD 34

// ---------------------------------------------------------------------------
// WMMA GEMM:  C[M,64] = A[M,K] @ W^T   (+ optional bias[64])
//   W is [64,K] row-major, so B[k,n] = W[n,k].
//   Block = 256 threads = 8 waves; each wave computes a 16x64 output tile
//   (block tile: 128 rows x 64 cols). A/W staged through LDS in K-chunks of 32.
//   Uses V_WMMA_F32_16X16X4_F32 (fp32 in/out matches the reference; these
//   GEMMs are HBM-bandwidth-bound so fp32 matrix throughput is free).
//   A-tile global loads are non-temporal: v_feat/t_feat are streamed once and
//   must not evict the L2-resident graph working set (~37MB of 192MB L2).
// ---------------------------------------------------------------------------
__global__ __launch_bounds__(256) void gemm_wmma_f32(
    const float* __restrict__ A,     // [M,K]
    const float* __restrict__ W,     // [64,K]
    const float* __restrict__ bias,  // [64] or nullptr
    float* __restrict__ C,           // [M,64]
    int M, int K)
{
    __shared__ float sA[128][LPAD];
    __shared__ float sW[64][LPAD];

    const int tid  = threadIdx.x;
    const int wave = tid >> 5;
    const int lane = tid & 31;
    const int rowBlock = blockIdx.x * 128;

    v8f acc0 = {}, acc1 = {}, acc2 = {}, acc3 = {};

    const int arow = wave * 16 + (lane & 15);
    const int kb   = (lane >> 4) * 2;     // lanes 0-15: K {0,1}; lanes 16-31: K {2,3}
    const int nlo  = lane & 31 & 15;

    for (int k0 = 0; k0 < K; k0 += 32) {
        // cooperative load: 128 rows x 32 cols of A (1024 16B slots), NT
        #pragma unroll
        for (int i = 0; i < 4; ++i) {
            int slot = i * 256 + tid;                 // 0..1023
            int r  = slot >> 3;
            int c4 = (slot & 7) << 2;
            int gr = rowBlock + r;
            v4f v = {0.f, 0.f, 0.f, 0.f};
            if (gr < M)
                v = __builtin_nontemporal_load(
                        (const v4f*)(A + (size_t)gr * K + k0 + c4));
            v2f lo, hi; lo.x = v.x; lo.y = v.y; hi.x = v.z; hi.y = v.w;
            *(v2f*)&sA[r][c4 + 0] = lo;               // 8B-aligned ds_store_b64
            *(v2f*)&sA[r][c4 + 2] = hi;
        }
        // cooperative load: 64 rows x 32 cols of W (512 16B slots), temporal
        #pragma unroll
        for (int i = 0; i < 2; ++i) {
            int slot = i * 256 + tid;                 // 0..511
            int r  = slot >> 3;
            int c4 = (slot & 7) << 2;
            v4f v = *(const v4f*)(W + (size_t)r * K + k0 + c4);
            v2f lo, hi; lo.x = v.x; lo.y = v.y; hi.x = v.z; hi.y = v.w;
            *(v2f*)&sW[r][c4 + 0] = lo;
            *(v2f*)&sW[r][c4 + 2] = hi;
        }
        __syncthreads();

        #pragma unroll
        for (int kk = 0; kk < 32; kk += 4) {
            // A fragment (16x4 f32): lane<16 -> row=lane, K={kk,kk+1};
            //                        lane>=16 -> row=lane-16, K={kk+2,kk+3}
            v2f a  = *(const v2f*)&sA[arow][kk + kb];       // ds_load_b64
            // B fragments (4x16): B[k][n] = W[n][k], mirrored lane layout
            v2f b0 = *(const v2f*)&sW[ 0 + nlo][kk + kb];
            v2f b1 = *(const v2f*)&sW[16 + nlo][kk + kb];
            v2f b2 = *(const v2f*)&sW[32 + nlo][kk + kb];
            v2f b3 = *(const v2f*)&sW[48 + nlo][kk + kb];

            acc0 = __builtin_amdgcn_wmma_f32_16x16x4_f32(false, a, false, b0,
                                                         (short)0, acc0, false, false);
            acc1 = __builtin_amdgcn_wmma_f32_16x16x4_f32(false, a, false, b1,
                                                         (short)0, acc1, false, false);
            acc2 = __builtin_amdgcn_wmma_f32_16x16x4_f32(false, a, false, b2,
                                                         (short)0, acc2, false, false);
            acc3 = __builtin_amdgcn_wmma_f32_16x16x4_f32(false, a, false, b3,
                                                         (short)0, acc3, false, false);
        }
        __syncthreads();
    }

    // Store. C/D f32 16x16 layout: VGPR r, lanes 0-15: M=r, N=lane;
    //                               lanes 16-31: M=8+r, N=lane-16.
    const int mBase = rowBlock + wave * 16 + ((lane >> 4) << 3);
    float bia0 = bias ? bias[ 0 + nlo] : 0.0f;
    float bia1 = bias ? bias[16 + nlo] : 0.0f;
    float bia2 = bias ? bias[32 + nlo] : 0.0f;
    float bia3 = bias ? bias[48 + nlo] : 0.0f;
    #pragma unroll
    for (int r = 0; r < 8; ++r) {
        int m = mBase + r;
        if (m < M) {
            float* row = C + (size_t)m * 64;
            row[ 0 + nlo] = acc0[r] + bia0;
            row[16 + nlo] = acc1[r] + bia1;
            row[32 + nlo] = acc2[r] + bia2;
            row[48 + nlo] = acc3[r] + bia3;
        }
    }
}

// ---------------------------------------------------------------------------
// Degree accumulation: deg[dst[e]] += 1  (2M atomics, L2-resident)
// ---------------------------------------------------------------------------
__global__ void deg_kernel(const int* __restrict__ dst, float* __restrict__ deg, int E)
{
    int e = blockIdx.x * blockDim.x + threadIdx.x;
    if (e < E) atomicAdd(&deg[dst[e]], 1.0f);
}

__global__ void dinv_kernel(float* __restrict__ deg_to_dinv, int n)
{
    int i = blockIdx.x * blockDim.x + threadIdx.x;
    if (i < n) deg_to_dinv[i] = rsqrtf(deg_to_dinv[i] + 1.0f);
}

// ---------------------------------------------------------------------------
// x0 = concat(user_emb, item_part); also seed the running layer-sum (in d_out)
// ---------------------------------------------------------------------------
__global__ void init_x0_kernel(const float* __restrict__ user_emb,
                               const float* __restrict__ item_part,
                               float* __restrict__ x0,
                               float* __restrict__ sum_out)
{
    int idx = blockIdx.x * blockDim.x + threadIdx.x;
    const int total = N_CNT * DE;
    if (idx >= total) return;
    float v = (idx < U_CNT * DE) ? user_emb[idx] : item_part[idx - U_CNT * DE];
    x0[idx] = v;
    sum_out[idx] = v;
}

// ---------------------------------------------------------------------------
// Edge propagation: out[dst] += h[src] * dinv[src]*dinv[dst]
//   one thread per (edge, channel); 64 channels of an edge are 256B contiguous
//   h (20.5MB) is L2-resident on the 192MB L2, so gather + atomics stay in L2.
// ---------------------------------------------------------------------------
__global__ void edge_kernel(const float* __restrict__ h,
                            const int* __restrict__ src,
                            const int* __restrict__ dst,
                            const float* __restrict__ dinv,
                            float* __restrict__ out, int E)
{
    long long idx = (long long)blockIdx.x * blockDim.x + threadIdx.x;
    long long total = (long long)E * DE;
    if (idx >= total) return;
    int e = (int)(idx >> 6);
    int c = (int)(idx & 63);
    int s = src[e];
    int d = dst[e];
    float w = dinv[s] * dinv[d];
    atomicAdd(&out[(size_t)d * DE + c], h[(size_t)s * DE + c] * w);
}

// ---------------------------------------------------------------------------
// Self-loop + bias + running layer sum:
//   x[idx] += h[idx]*dinv[i]^2 + b[c];  sum[idx] += x[idx]
// ---------------------------------------------------------------------------
__global__ void finalize_kernel(float* __restrict__ x,
                                const float* __restrict__ h,
                                const float* __restrict__ dinv,
                                const float* __restrict__ b,
                                float* __restrict__ sum_out)
{
    int idx = blockIdx.x * blockDim.x + threadIdx.x;
    const int total = N_CNT * DE;
    if (idx >= total) return;
    int i = idx >> 6;
    int c = idx & 63;
    float di = dinv[i];
    float v = x[idx] + h[idx] * di * di + b[c];
    x[idx] = v;
    sum_out[idx] += v;
}

__global__ void scale_kernel(float* __restrict__ p, long long n, float s)
{
    long long idx = (long long)blockIdx.x * blockDim.x + threadIdx.x;
    if (idx < n) p[idx] *= s;
}

// ---------------------------------------------------------------------------
// Host-side driver
// ---------------------------------------------------------------------------
static void run_stack(const float* user_emb, const float* item_part,
                      const float* Ws, const float* bs,       // [3,64,64], [3,64]
                      const int* esrc, const int* edst, int E2,
                      const float* dinv,
                      float* xa, float* xb, float* xc,        // N*64 scratch each
                      float* sum_out,                          // slice of d_out
                      hipStream_t stream)
{
    const int totalN   = N_CNT * DE;
    const int blkElem  = (totalN + 255) / 256;
    const long long totE = (long long)E2 * DE;
    const int blkEdge  = (int)((totE + 255) / 256);
    const int blkGemm  = (N_CNT + 127) / 128;

    init_x0_kernel<<<blkElem, 256, 0, stream>>>(user_emb, item_part, xa, sum_out);

    float* cur = xa;
    float* nxt = xc;
    for (int l = 0; l < 3; ++l) {
        // h = x @ W_l^T
        gemm_wmma_f32<<<blkGemm, 256, 0, stream>>>(cur, Ws + (size_t)l * DE * DE,
                                                   nullptr, xb, N_CNT, DE);
        (void)hipMemsetAsync(nxt, 0, (size_t)totalN * sizeof(float), stream);
        edge_kernel<<<blkEdge, 256, 0, stream>>>(xb, esrc, edst, dinv, nxt, E2);
        finalize_kernel<<<blkElem, 256, 0, stream>>>(nxt, xb, dinv,
                                                     bs + (size_t)l * DE, sum_out);
        float* t = cur; cur = nxt; nxt = t;
    }
}

extern "C" void kernel_launch(void* const* d_in, const int* in_sizes, int n_in,
                              void* d_out, int out_size, void* d_ws, size_t ws_size,
                              hipStream_t stream)
{
    const float* user_emb = (const float*)d_in[0];
    const float* item_emb = (const float*)d_in[1];
    const float* v_feat   = (const float*)d_in[2];   // [I, 4096]
    const float* t_feat   = (const float*)d_in[3];   // [I, 1024]
    const float* Wv       = (const float*)d_in[4];   // [64, 4096]
    const float* bv       = (const float*)d_in[5];
    const float* Wt       = (const float*)d_in[6];   // [64, 1024]
    const float* bt       = (const float*)d_in[7];
    const float* Wg       = (const float*)d_in[8];   // [3,64,64]
    const float* bg       = (const float*)d_in[9];   // [3,64]
    const float* Wm       = (const float*)d_in[10];  // [3,64,64]
    const float* bm       = (const float*)d_in[11];  // [3,64]
    const int*   esrc     = (const int*)d_in[12];
    const int*   edst     = (const int*)d_in[13];
    const int    E2       = in_sizes[12];            // 2,000,000

    // workspace layout (floats)
    float* ws      = (float*)d_ws;
    float* dinv    = ws;                                  // N
    float* visual  = dinv   + N_CNT;                      // I*64
    float* textual = visual + (size_t)I_CNT * DE;         // I*64
    float* xa      = textual+ (size_t)I_CNT * DE;         // N*64
    float* xb      = xa     + (size_t)N_CNT * DE;         // N*64
    float* xc      = xb     + (size_t)N_CNT * DE;         // N*64

    float* out_g = (float*)d_out;                         // [N,64]
    float* out_v = out_g + (size_t)N_CNT * DE;            // [N,64]
    float* out_t = out_v + (size_t)N_CNT * DE;            // [N,64]

    // 1) degrees -> dinv
    (void)hipMemsetAsync(dinv, 0, (size_t)N_CNT * sizeof(float), stream);
    deg_kernel<<<(E2 + 255) / 256, 256, 0, stream>>>(edst, dinv, E2);
    dinv_kernel<<<(N_CNT + 255) / 256, 256, 0, stream>>>(dinv, N_CNT);

    // 2) modality feature transforms (HBM-bound WMMA GEMMs)
    gemm_wmma_f32<<<(I_CNT + 127) / 128, 256, 0, stream>>>(v_feat, Wv, bv, visual,
                                                           I_CNT, 4096);
    gemm_wmma_f32<<<(I_CNT + 127) / 128, 256, 0, stream>>>(t_feat, Wt, bt, textual,
                                                           I_CNT, 1024);

    // 3) three GCN stacks; layer-means accumulate straight into d_out slices
    run_stack(user_emb, item_emb, Wg, bg, esrc, edst, E2, dinv,
              xa, xb, xc, out_g, stream);
    run_stack(user_emb, visual,   Wm, bm, esrc, edst, E2, dinv,
              xa, xb, xc, out_v, stream);
    run_stack(user_emb, textual,  Wm, bm, esrc, edst, E2, dinv,
              xa, xb, xc, out_t, stream);

    // 4) mean over 4 layer snapshots
    long long totOut = 3LL * N_CNT * DE;
    scale_kernel<<<(int)((totOut + 255) / 256), 256, 0, stream>>>((float*)d_out,
                                                                  totOut, 0.25f);
}